// GNN_80341658239112
// MI455X (gfx1250) — compile-verified
//
#include <hip/hip_runtime.h>

typedef __attribute__((ext_vector_type(2))) float v2f;
typedef __attribute__((ext_vector_type(8))) float v8f;

#define BATCH 4
#define CH    256
#define WH    4096

// ---- WMMA GEMM tiling ----
// block tile 64(M) x 128(N), 4 waves in 2x2, each wave 32x64 = 2x4 WMMA tiles
#define BM   64
#define BN   128
#define KB   32
#define LDT  (KB + 8)    // padded [row][k] stride (floats); 160B rows, 16B aligned
#define BNP  (BN + 4)    // padded [k][n] stride (floats);  528B rows, 16B aligned
#define TSZA (BM * LDT)  // 2560 floats
#define TSZB (BN * LDT)  // 5120 floats >= KB*BNP (4224)

// async copy 16B global -> LDS (ASYNCcnt); LDS offset = low 32 bits of generic ptr
__device__ __forceinline__ void async_copy_b128(const float* g, const float* l)
{
    unsigned loff = (unsigned)(unsigned long long)l;
    asm volatile("global_load_async_to_lds_b128 %0, %1, off"
                 :: "v"(loff), "v"(g) : "memory");
}

__device__ __forceinline__ void wait_async0()
{
#if __has_builtin(__builtin_amdgcn_s_wait_asynccnt)
    __builtin_amdgcn_s_wait_asynccnt(0);
#else
    asm volatile("s_wait_asynccnt 0x0" ::: "memory");
#endif
}

// C[M x N] = A[M x K] * op(B) (+ epilogue)
//   BT=true : B is [N x K] row-major (C = A * B^T)   -- logits
//   BT=false: B is [K x N] row-major (C = A * B)     -- linears / attention apply
// EPI: 0 = store; 1 = relu(x + bias[n]); 2 = x + add1[idx] + add2[idx]
template<bool BT, int EPI>
__global__ __launch_bounds__(128)
void wmma_gemm_f32(const float* __restrict__ A, const float* __restrict__ Bm,
                   float* __restrict__ Cm, int M, int N, int K,
                   const float* __restrict__ bias,
                   const float* __restrict__ add1,
                   const float* __restrict__ add2,
                   long sA, long sB, long sC)
{
    __shared__ __align__(16) float At[2][TSZA];
    __shared__ __align__(16) float Bt[2][TSZB];

    const int tid  = threadIdx.x;        // 0..127, 4 waves (wave32)
    const int wave = tid >> 5;
    const int lane = tid & 31;
    const int l15  = lane & 15;
    const int lhi  = lane >> 4;          // half-wave select
    const int wm   = (wave >> 1) << 5;   // wave 2x2 grid -> each wave 32(M) x 64(N)
    const int wn   = (wave & 1) << 6;
    const int m0   = blockIdx.y * BM;
    const int n0   = blockIdx.x * BN;
    const long bz  = blockIdx.z;

    A  += bz * sA;
    Bm += bz * sB;
    Cm += bz * sC;

    v8f acc[2][4] = {};

    // issue one k-chunk of async global->LDS copies into buffer b
    auto fill = [&](int b, int kc) {
        #pragma unroll
        for (int it = 0; it < 4; ++it) {            // A: 64 rows x 32 k = 512 x 16B
            int slot = tid + it * 128;
            int r    = slot >> 3;
            int kq   = (slot & 7) << 2;
            async_copy_b128(A + (long)(m0 + r) * K + kc + kq, &At[b][r * LDT + kq]);
        }
        if (BT) {
            // B rows k-contiguous: Bt[n][k], 128 rows x 32 k = 1024 x 16B
            #pragma unroll
            for (int it = 0; it < 8; ++it) {
                int slot = tid + it * 128;
                int r    = slot >> 3;
                int kq   = (slot & 7) << 2;
                async_copy_b128(Bm + (long)(n0 + r) * K + kc + kq, &Bt[b][r * LDT + kq]);
            }
        } else {
            // B is [K][N]: keep untransposed in LDS, Bt[k][n] (async is byte-exact)
            #pragma unroll
            for (int it = 0; it < 8; ++it) {        // 32 k x 32 n-quads
                int slot = tid + it * 128;
                int k    = slot >> 5;
                int nq   = (slot & 31) << 2;
                async_copy_b128(Bm + (long)(kc + k) * N + n0 + nq, &Bt[b][k * BNP + nq]);
            }
        }
    };

    int buf = 0;
    fill(0, 0);
    wait_async0();
    __syncthreads();

    for (int kc = 0; kc < K; kc += KB) {
        if (kc + KB < K) fill(buf ^ 1, kc + KB);   // prefetch overlaps WMMAs

        const float* ap = At[buf];
        const float* bp = Bt[buf];
        #pragma unroll
        for (int kk = 0; kk < KB; kk += 4) {
            const int kf = kk + (lhi << 1);        // lanes16-31 hold K+2,K+3 (ISA 7.12.2)
            v2f a[2], b[4];
            #pragma unroll
            for (int tm = 0; tm < 2; ++tm)
                a[tm] = *(const v2f*)(ap + (wm + tm * 16 + l15) * LDT + kf);
            #pragma unroll
            for (int tn = 0; tn < 4; ++tn) {
                if (BT) {
                    b[tn] = *(const v2f*)(bp + (wn + tn * 16 + l15) * LDT + kf);
                } else {
                    b[tn].x = bp[(kf    ) * BNP + wn + tn * 16 + l15];
                    b[tn].y = bp[(kf + 1) * BNP + wn + tn * 16 + l15];
                }
            }
            #pragma unroll
            for (int tm = 0; tm < 2; ++tm)
                #pragma unroll
                for (int tn = 0; tn < 4; ++tn)
                    acc[tm][tn] = __builtin_amdgcn_wmma_f32_16x16x4_f32(
                        false, a[tm], false, b[tn], (short)0, acc[tm][tn], false, false);
        }

        wait_async0();     // next buffer landed in LDS
        __syncthreads();   // everyone done reading current buffer
        buf ^= 1;
    }

    // epilogue: C layout (ISA 7.12.2): VGPR i -> M = i + 8*lhi ; lane -> N
    #pragma unroll
    for (int tm = 0; tm < 2; ++tm) {
        #pragma unroll
        for (int tn = 0; tn < 4; ++tn) {
            const int nn = n0 + wn + tn * 16 + l15;
            #pragma unroll
            for (int i = 0; i < 8; ++i) {
                const int mm = m0 + wm + tm * 16 + i + (lhi << 3);
                float v = acc[tm][tn][i];
                const long idx = (long)mm * N + nn;
                if (EPI == 1) v = fmaxf(v + bias[nn], 0.0f);
                if (EPI == 2) v = v + add1[idx] + add2[idx];
                Cm[idx] = v;
            }
        }
    }
}

// (b, C, WH) -> (b, WH, C)
__global__ __launch_bounds__(256)
void transpose_ker(const float* __restrict__ in, float* __restrict__ out)
{
    __shared__ float tile[32][33];
    const int b  = blockIdx.z;
    const int c0 = blockIdx.y * 32;
    const int n0 = blockIdx.x * 32;
    const float* ip = in  + (long)b * CH * WH;
    float*       op = out + (long)b * WH * CH;
    const int tx = threadIdx.x, ty = threadIdx.y;   // (32, 8)
    #pragma unroll
    for (int i = 0; i < 32; i += 8)
        tile[ty + i][tx] = ip[(long)(c0 + ty + i) * WH + n0 + tx];
    __syncthreads();
    #pragma unroll
    for (int i = 0; i < 32; i += 8)
        op[(long)(n0 + ty + i) * CH + c0 + tx] = tile[tx][ty + i];
}

// in-place row softmax over a (WH x WH) matrix; one block per row
__global__ __launch_bounds__(256)
void softmax_rows_ker(float* __restrict__ G)
{
    const int t = threadIdx.x;
    float4* row = (float4*)(G + (long)blockIdx.x * WH);
    __shared__ float red[256];
    float4 v[4];
    float mx = -3.402823e38f;
    #pragma unroll
    for (int i = 0; i < 4; ++i) {
        v[i] = row[i * 256 + t];
        mx = fmaxf(mx, fmaxf(fmaxf(v[i].x, v[i].y), fmaxf(v[i].z, v[i].w)));
    }
    red[t] = mx; __syncthreads();
    for (int s = 128; s > 0; s >>= 1) {
        if (t < s) red[t] = fmaxf(red[t], red[t + s]);
        __syncthreads();
    }
    mx = red[0]; __syncthreads();
    float sum = 0.0f;
    #pragma unroll
    for (int i = 0; i < 4; ++i) {
        v[i].x = __expf(v[i].x - mx); v[i].y = __expf(v[i].y - mx);
        v[i].z = __expf(v[i].z - mx); v[i].w = __expf(v[i].w - mx);
        sum += v[i].x + v[i].y + v[i].z + v[i].w;
    }
    red[t] = sum; __syncthreads();
    for (int s = 128; s > 0; s >>= 1) {
        if (t < s) red[t] += red[t + s];
        __syncthreads();
    }
    const float inv = 1.0f / red[0];
    #pragma unroll
    for (int i = 0; i < 4; ++i) {
        v[i].x *= inv; v[i].y *= inv; v[i].z *= inv; v[i].w *= inv;
        row[i * 256 + t] = v[i];
    }
}

// two-phase column sum (deterministic: no float atomics)
#define CSEG 16
__global__ __launch_bounds__(256)
void colsum_part_ker(const float* __restrict__ G, float* __restrict__ part)
{
    const int m  = blockIdx.x * 256 + threadIdx.x;
    const int n0 = blockIdx.y * (WH / CSEG);
    float s = 0.0f;
    for (int n = 0; n < WH / CSEG; ++n) s += G[(long)(n0 + n) * WH + m];
    part[(long)blockIdx.y * WH + m] = s;
}

__global__ __launch_bounds__(256)
void colsum_finish_ker(const float* __restrict__ part, float* __restrict__ invcs)
{
    const int m = blockIdx.x * 256 + threadIdx.x;
    float s = 0.0f;
    #pragma unroll
    for (int seg = 0; seg < CSEG; ++seg) s += part[(long)seg * WH + m];
    invcs[m] = 1.0f / fmaxf(s, 1e-12f);   // jnp.clip(sum, EPS)
}

// X[m][d] *= invcs[m]   (folds the column-L1 normalization into ax)
__global__ __launch_bounds__(256)
void scale_rows_ker(float* __restrict__ X, const float* __restrict__ invcs)
{
    const long id = (long)blockIdx.x * 256 + threadIdx.x;   // float4 index
    float4* p = (float4*)X;
    float4 v = p[id];
    const float s = invcs[(int)(id >> 6)];                  // CH/4 = 64 quads per row
    v.x *= s; v.y *= s; v.z *= s; v.w *= s;
    p[id] = v;
}

// one graph stage: out = A_norm @ relu(R@Wa+ba) + relu(S@Wu+bu) + S
static void run_stage(const float* S, const float* R,
                      const float* Wa, const float* ba,
                      const float* Wu, const float* bu,
                      float* ax, float* ux, float* L,
                      float* invcs, float* csPart, float* outp,
                      hipStream_t stream)
{
    const long NB = (long)WH * CH;
    // ux = relu(S @ Wu + bu), all batches
    wmma_gemm_f32<false, 1><<<dim3(CH / BN, WH / BM, BATCH), 128, 0, stream>>>(
        S, Wu, ux, WH, CH, CH, bu, nullptr, nullptr, NB, 0, NB);
    // ax = relu(R @ Wa + ba), all batches
    wmma_gemm_f32<false, 1><<<dim3(CH / BN, WH / BM, BATCH), 128, 0, stream>>>(
        R, Wa, ax, WH, CH, CH, ba, nullptr, nullptr, NB, 0, NB);

    for (int i = 0; i < BATCH; ++i) {
        const float* Si = S + (long)i * NB;
        const float* Ri = R + (long)i * NB;
        // L = Si @ Ri^T  (4096 x 4096, K=256)
        wmma_gemm_f32<true, 0><<<dim3(WH / BN, WH / BM, 1), 128, 0, stream>>>(
            Si, Ri, L, WH, WH, CH, nullptr, nullptr, nullptr, 0, 0, 0);
        softmax_rows_ker<<<WH, 256, 0, stream>>>(L);
        colsum_part_ker<<<dim3(WH / 256, CSEG), 256, 0, stream>>>(L, csPart);
        colsum_finish_ker<<<WH / 256, 256, 0, stream>>>(csPart, invcs);
        scale_rows_ker<<<(WH * CH / 4) / 256, 256, 0, stream>>>(ax + (long)i * NB, invcs);
        // out_i = L @ ax_i + ux_i + Si  (4096 x 256, K=4096)
        wmma_gemm_f32<false, 2><<<dim3(CH / BN, WH / BM, 1), 128, 0, stream>>>(
            L, ax + (long)i * NB, outp + (long)i * NB, WH, CH, WH,
            nullptr, ux + (long)i * NB, Si, 0, 0, 0);
    }
}

extern "C" void kernel_launch(void* const* d_in, const int* in_sizes, int n_in,
                              void* d_out, int out_size, void* d_ws, size_t ws_size,
                              hipStream_t stream)
{
    (void)in_sizes; (void)n_in; (void)out_size; (void)ws_size;
    const float* skt = (const float*)d_in[0];
    const float* ref = (const float*)d_in[1];
    const float* Wa3 = (const float*)d_in[2];
    const float* ba3 = (const float*)d_in[3];
    const float* Wu3 = (const float*)d_in[4];
    const float* bu3 = (const float*)d_in[5];
    const float* Wa4 = (const float*)d_in[6];
    const float* ba4 = (const float*)d_in[7];
    const float* Wu4 = (const float*)d_in[8];
    const float* bu4 = (const float*)d_in[9];
    float* out = (float*)d_out;

    const long NB = (long)WH * CH;        // per-batch (wh, c) elems
    float* Sf     = (float*)d_ws;                         // 4 * NB
    float* Rf     = Sf  + (long)BATCH * NB;               // 4 * NB
    float* gen    = Rf  + (long)BATCH * NB;               // 4 * NB
    float* ax     = gen + (long)BATCH * NB;               // 4 * NB
    float* ux     = ax  + (long)BATCH * NB;               // 4 * NB
    float* L      = ux  + (long)BATCH * NB;               // WH*WH (one batch)
    float* invcs  = L   + (long)WH * WH;                  // WH
    float* csPart = invcs + WH;                           // CSEG * WH

    dim3 tb(32, 8);
    transpose_ker<<<dim3(WH / 32, CH / 32, BATCH), tb, 0, stream>>>(skt, Sf);
    transpose_ker<<<dim3(WH / 32, CH / 32, BATCH), tb, 0, stream>>>(ref, Rf);

    // graph 1: gen = gconv(build_graph(Sf,Rf), Sf, Rf, W3) + Sf
    run_stage(Sf, Rf, Wa3, ba3, Wu3, bu3, ax, ux, L, invcs, csPart, gen, stream);
    // graph 2: out = gconv(build_graph(gen,gen), gen, gen, W4) + gen
    run_stage(gen, gen, Wa4, ba4, Wu4, bu4, ax, ux, L, invcs, csPart, out, stream);
}